// ExactMultitaskGP_81870666596361
// MI455X (gfx1250) — compile-verified
//
#include <hip/hip_runtime.h>
#include <hip/hip_bf16.h>

// ---------------------------------------------------------------------------
// Multitask exact GP on MI455X (gfx1250, wave32).
// Heavy math (Cholesky trailing updates, TRSM panel GEMMs, fvar SYRK) runs on
// V_WMMA_F32_16X16X4_F32. fp32 throughout (Cholesky chain needs it).
// The dominant SYRK double-buffers its LDS tiles with
// GLOBAL_LOAD_ASYNC_TO_LDS_B128 + s_wait_asynccnt (CDNA5 async copy path).
// ---------------------------------------------------------------------------

typedef __attribute__((ext_vector_type(2))) float v2f;
typedef __attribute__((ext_vector_type(8))) float v8f;

#define BLK 64                 // Cholesky / TRSM panel width
#define TKC 32                 // k-chunk depth for LDS-tiled kernels
#define SP  34                 // padded LDS stride (even for b64, conflict-free)

__device__ __forceinline__ v8f wmma_f32_4(v2f a, v2f b, v8f c) {
  // D = A(16x4) * B(4x16) + C(16x16), fp32
  return __builtin_amdgcn_wmma_f32_16x16x4_f32(
      /*neg_a=*/false, a, /*neg_b=*/false, b,
      /*c_mod=*/(short)0, c, /*reuse_a=*/false, /*reuse_b=*/false);
}

// ---------------------------------------------------------------------------
// 1. task_K = covar_factor covar_factor^T + diag(exp(log_var))      (l x l)
// ---------------------------------------------------------------------------
__global__ void k_task(const float* __restrict__ cf, const float* __restrict__ log_var,
                       float* __restrict__ taskK, int l, int rank) {
  int idx = threadIdx.x;
  if (idx >= l * l) return;
  int i = idx / l, j = idx % l;
  float s = 0.f;
  for (int r = 0; r < rank; ++r) s += cf[i * rank + r] * cf[j * rank + r];
  if (i == j) s += __expf(log_var[i]);
  taskK[idx] = s;
}

// ---------------------------------------------------------------------------
// 2. RBF kernel matrix: out[i,j] = exp(-0.5 * ||x1_i - x2_j||^2 / ls^2)
// ---------------------------------------------------------------------------
__global__ void k_rbf(const float* __restrict__ X1, const float* __restrict__ X2,
                      float* __restrict__ out, int n1, int n2, int D,
                      const float* __restrict__ log_ls) {
  int j = blockIdx.x * blockDim.x + threadIdx.x;
  int i = blockIdx.y;
  if (j >= n2) return;
  float ls2 = __expf(2.0f * log_ls[0]);
  float d2 = 0.f;
  for (int d = 0; d < D; ++d) {
    float df = X1[i * D + d] - X2[j * D + d];
    d2 += df * df;
  }
  out[(size_t)i * n2 + j] = __expf(-0.5f * d2 / ls2);
}

// ---------------------------------------------------------------------------
// 3a. K = kron(task_K, rbf_train) + kron(diag(exp(log_noise)), I);
//     also writes the `noise` output matrix (dense, mostly zeros).
// ---------------------------------------------------------------------------
__global__ void k_assemble_K(const float* __restrict__ taskK, const float* __restrict__ rbf,
                             const float* __restrict__ log_noise,
                             float* __restrict__ K, float* __restrict__ noise_out,
                             int N, int l) {
  int M = N * l;
  int col = blockIdx.x * blockDim.x + threadIdx.x;
  int row = blockIdx.y;
  if (col >= M) return;
  int t1 = row / N, i = row % N;
  int t2 = col / N, j = col % N;
  float v = taskK[t1 * l + t2] * rbf[(size_t)i * N + j];
  float nz = 0.f;
  if (row == col) {
    float e = __expf(log_noise[t1]);
    v += e;
    nz = e;
  }
  K[(size_t)row * M + col] = v;
  noise_out[(size_t)row * M + col] = nz;
}

// 3b. Generic kron(task_K (l x l), rbf (Nr x Nc)) -> out ((Nr*l) x (Nc*l))
__global__ void k_kron(const float* __restrict__ taskK, const float* __restrict__ rbf,
                       float* __restrict__ out, int Nr, int Nc, int l) {
  int cols = Nc * l;
  int col = blockIdx.x * blockDim.x + threadIdx.x;
  int row = blockIdx.y;
  if (col >= cols) return;
  int t1 = row / Nr, i = row % Nr;
  int t2 = col / Nc, j = col % Nc;
  out[(size_t)row * cols + col] = taskK[t1 * l + t2] * rbf[(size_t)i * Nc + j];
}

// ---------------------------------------------------------------------------
// 4. Cholesky, blocked (BLK=64), right-looking. Lower triangle of K -> L.
// ---------------------------------------------------------------------------

// 4a. Factor 64x64 diagonal block in LDS (one workgroup of 64 threads).
__global__ void k_potrf(float* __restrict__ K, int M, int kb) {
  __shared__ float s[BLK * BLK];          // 16 KB
  float* blk = K + (size_t)(kb * BLK) * M + (size_t)kb * BLK;
  int t = threadIdx.x;                    // 0..63, column index on load
  for (int r = 0; r < BLK; ++r) s[r * BLK + t] = blk[(size_t)r * M + t];
  __syncthreads();
  for (int k = 0; k < BLK; ++k) {
    float dkk = s[k * BLK + k];
    __syncthreads();
    float d = sqrtf(dkk);
    if (t == k) s[k * BLK + k] = d;
    else if (t > k) s[t * BLK + k] = s[t * BLK + k] / d;
    __syncthreads();
    if (t > k) {
      float lik = s[t * BLK + k];
      for (int j = k + 1; j <= t; ++j) s[t * BLK + j] -= lik * s[j * BLK + k];
    }
    __syncthreads();
  }
  for (int r = 0; r < BLK; ++r)
    blk[(size_t)r * M + t] = (t <= r) ? s[r * BLK + t] : 0.0f;
}

// 4b. Panel solve: for row-block ib>kb: X * Lii^T = K[ib,kb]; X overwrites.
//     Rows are independent -> one thread per row, no syncs in the sweep.
__global__ void k_trsm_right(float* __restrict__ K, int M, int kb) {
  __shared__ float sL[BLK * BLK];         // 16 KB
  __shared__ float sB[BLK * (BLK + 1)];   // padded rows: bank-conflict free
  int ib = kb + 1 + blockIdx.x;
  int t = threadIdx.x;                    // 0..63
  const float* Lg = K + (size_t)(kb * BLK) * M + (size_t)kb * BLK;
  float* Bg = K + (size_t)(ib * BLK) * M + (size_t)kb * BLK;
  for (int r = 0; r < BLK; ++r) {
    sL[r * BLK + t] = Lg[(size_t)r * M + t];
    sB[r * (BLK + 1) + t] = Bg[(size_t)r * M + t];
  }
  __syncthreads();
  float* x = &sB[t * (BLK + 1)];          // this thread's row
  for (int j = 0; j < BLK; ++j) {
    float xv = x[j] / sL[j * BLK + j];
    x[j] = xv;
    for (int j2 = j + 1; j2 < BLK; ++j2) x[j2] -= xv * sL[j2 * BLK + j];
  }
  __syncthreads();
  for (int r = 0; r < BLK; ++r) Bg[(size_t)r * M + t] = sB[r * (BLK + 1) + t];
}

// 4c. Trailing update (WMMA): K[ib,jb] -= P_ib * P_jb^T, P = column panel kb.
//     grid = (2, #pairs), 256 threads = 8 waves, one 16x16 tile per wave.
//     Both operands are contiguous b64 loads from the L2-hot 64-wide panel.
__global__ void k_syrk_chol(float* __restrict__ K, int M, int kb, int nb) {
  int p = blockIdx.y;
  int a = 0;
  while ((a + 1) * (a + 2) / 2 <= p) ++a;       // triangular decode
  int b = p - a * (a + 1) / 2;
  int ib = kb + 1 + a, jb = kb + 1 + b;

  int wave = threadIdx.x >> 5;
  int lane = threadIdx.x & 31;
  int ti = blockIdx.x * 8 + wave;               // 0..15 tiles in 64x64 block
  int tr = ti >> 2, tc = ti & 3;

  int row0 = ib * BLK + tr * 16;
  int col0 = jb * BLK + tc * 16;
  int m  = lane & 15;
  int ko = (lane >> 4) << 1;                    // 0 or 2

  const float* Arow = K + (size_t)(row0 + m) * M + (size_t)kb * BLK;  // P_ib row
  const float* Brow = K + (size_t)(col0 + m) * M + (size_t)kb * BLK;  // P_jb row

  v8f acc = {};
#pragma unroll 4
  for (int kk = 0; kk < BLK; kk += 4) {
    v2f av = *(const v2f*)&Arow[kk + ko];
    v2f bv = *(const v2f*)&Brow[kk + ko];
    acc = wmma_f32_4(av, bv, acc);
  }
  int rbase = row0 + ((lane >> 4) << 3);
  int col = col0 + m;
#pragma unroll
  for (int v = 0; v < 8; ++v)
    K[(size_t)(rbase + v) * M + col] -= acc[v];
}

// ---------------------------------------------------------------------------
// 5. Blocked forward TRSM: A = L^{-1} Kx (in place on Kx), 64-row panels.
// ---------------------------------------------------------------------------

// 5a. WMMA GEMM update: C(64 x Nc) -= A(64 x klen) * B(klen x Nc).
//     WG = 256 threads covers a 64x128 C block (4 tile-rows x 8 tile-waves).
//     B strip staged to LDS transposed [col][k] -> ds_load_b64 operands.
__global__ void k_gemm_nn_sub(float* __restrict__ C, int ldc,
                              const float* __restrict__ A, int lda,
                              const float* __restrict__ B, int ldb, int klen) {
  __shared__ float sB[128 * SP];              // 17 KB
  int tid = threadIdx.x;
  int wave = tid >> 5;
  int lane = tid & 31;
  int col0 = blockIdx.x * 128;
  int m  = lane & 15;
  int ko = (lane >> 4) << 1;

  v8f acc[4];
#pragma unroll
  for (int t = 0; t < 4; ++t) acc[t] = (v8f){};

  for (int kk = 0; kk < klen; kk += TKC) {
    __syncthreads();
    for (int idx = tid; idx < TKC * 128; idx += 256) {
      int kr = idx >> 7, c = idx & 127;
      sB[c * SP + kr] = B[(size_t)(kk + kr) * ldb + col0 + c];
      if (kk + TKC < klen)
        __builtin_prefetch(&B[(size_t)(kk + TKC + kr) * ldb + col0 + c], 0, 1);
    }
    __syncthreads();
#pragma unroll
    for (int k4 = 0; k4 < TKC; k4 += 4) {
      v2f bv = *(const v2f*)&sB[(wave * 16 + m) * SP + k4 + ko];
#pragma unroll
      for (int tr = 0; tr < 4; ++tr) {
        v2f av = *(const v2f*)&A[(size_t)(tr * 16 + m) * lda + kk + k4 + ko];
        acc[tr] = wmma_f32_4(av, bv, acc[tr]);
      }
    }
  }
  int col = col0 + wave * 16 + m;
  int rsel = (lane >> 4) << 3;
#pragma unroll
  for (int tr = 0; tr < 4; ++tr) {
    int rbase = tr * 16 + rsel;
#pragma unroll
    for (int v = 0; v < 8; ++v)
      C[(size_t)(rbase + v) * ldc + col] -= acc[tr][v];
  }
}

// 5b. Diagonal-block solve: L_ii X = Bpanel (64 x Mt), columns independent.
//     grid = Mt/128, 128 threads; one column per thread, no syncs in sweep.
__global__ void k_trsm_left(const float* __restrict__ K, int M, int ib,
                            float* __restrict__ Bmat, int Mt) {
  __shared__ float sL[BLK * BLK];             // 16 KB
  __shared__ float sB[BLK * 130];             // 33 KB, stride 130 vs banks
  int t = threadIdx.x;                        // 0..127
  int col0 = blockIdx.x * 128;
  const float* Lg = K + (size_t)(ib * BLK) * M + (size_t)ib * BLK;
  float* Bg = Bmat + (size_t)(ib * BLK) * Mt;

  for (int idx = t; idx < BLK * BLK; idx += 128) sL[idx] = Lg[(size_t)(idx >> 6) * M + (idx & 63)];
  for (int k = 0; k < BLK; ++k) sB[k * 130 + t] = Bg[(size_t)k * Mt + col0 + t];
  __syncthreads();
  for (int k = 0; k < BLK; ++k) {
    float x = sB[k * 130 + t] / sL[k * BLK + k];
    sB[k * 130 + t] = x;
    for (int k2 = k + 1; k2 < BLK; ++k2) sB[k2 * 130 + t] -= sL[k2 * BLK + k] * x;
  }
  __syncthreads();
  for (int k = 0; k < BLK; ++k) Bg[(size_t)k * Mt + col0 + t] = sB[k * 130 + t];
}

// ---------------------------------------------------------------------------
// 6. fvar SYRK (WMMA): C(Mt x Mt) -= A^T A, A is (M x Mt).
//    WG = 256 threads computes a 128x128 C macro-tile. Operand strips are
//    DOUBLE-BUFFERED in LDS via GLOBAL_LOAD_ASYNC_TO_LDS_B128: the next
//    32x128 chunk streams memory->LDS (no VGPR round trip) while WMMAs
//    consume the current one. In-order async completion => after issuing 8
//    new per-wave transfers, s_wait_asynccnt 8 retires the previous chunk;
//    a workgroup barrier then publishes it WG-wide.
// ---------------------------------------------------------------------------
__global__ void k_syrk_at_a(float* __restrict__ C, const float* __restrict__ A,
                            int M, int Mt) {
  __shared__ __align__(16) float sR[2][TKC * 128];   // rows operand, 2x16 KB
  __shared__ __align__(16) float sC[2][TKC * 128];   // cols operand, 2x16 KB
  int tid = threadIdx.x;
  int wave = tid >> 5;
  int lane = tid & 31;
  int row0 = blockIdx.y * 128;
  int col0 = blockIdx.x * 128;
  int m  = lane & 15;
  int ko = (lane >> 4) << 1;

  v8f acc[8];
#pragma unroll
  for (int t = 0; t < 8; ++t) acc[t] = (v8f){};

  // Issue one chunk's async transfers: 2048 b128 per buffer pair,
  // 4 loop iterations/thread x 2 issues = 8 async instructions per wave.
  auto stage = [&](int b, int kk) {
    for (int idx = tid; idx < TKC * 32; idx += 256) {
      int kr = idx >> 5;                 // k-row within chunk
      int cq = (idx & 31) << 2;          // 4-float column quad
      const float* gR = &A[(size_t)(kk + kr) * Mt + row0 + cq];
      const float* gC = &A[(size_t)(kk + kr) * Mt + col0 + cq];
      unsigned dR = (unsigned)(unsigned long long)(const void*)&sR[b][kr * 128 + cq];
      unsigned dC = (unsigned)(unsigned long long)(const void*)&sC[b][kr * 128 + cq];
      asm volatile("global_load_async_to_lds_b128 %0, %1, off"
                   :: "v"(dR), "v"(gR) : "memory");
      asm volatile("global_load_async_to_lds_b128 %0, %1, off"
                   :: "v"(dC), "v"(gC) : "memory");
    }
  };

  stage(0, 0);
  for (int kk = 0; kk < M; kk += TKC) {
    int cur = (kk / TKC) & 1;
    bool more = (kk + TKC) < M;
    __syncthreads();                       // retire all reads of buffer cur^1
    if (more) {
      stage(cur ^ 1, kk + TKC);            // prefetch next chunk (8 in flight)
      asm volatile("s_wait_asynccnt 0x8" ::: "memory");  // cur chunk landed
    } else {
      asm volatile("s_wait_asynccnt 0x0" ::: "memory");
    }
    __syncthreads();                       // publish cur chunk WG-wide

    const float* bR = &sR[cur][0];
    const float* bC = &sC[cur][0];
#pragma unroll
    for (int k4 = 0; k4 < TKC; k4 += 4) {
      v2f bv;
      bv.x = bC[(k4 + ko) * 128 + wave * 16 + m];
      bv.y = bC[(k4 + ko + 1) * 128 + wave * 16 + m];
#pragma unroll
      for (int tr = 0; tr < 8; ++tr) {
        v2f av;
        av.x = bR[(k4 + ko) * 128 + tr * 16 + m];
        av.y = bR[(k4 + ko + 1) * 128 + tr * 16 + m];
        acc[tr] = wmma_f32_4(av, bv, acc[tr]);
      }
    }
  }
  int col = col0 + wave * 16 + m;
  int rsel = (lane >> 4) << 3;
#pragma unroll
  for (int tr = 0; tr < 8; ++tr) {
    int rbase = row0 + tr * 16 + rsel;
#pragma unroll
    for (int v = 0; v < 8; ++v)
      C[(size_t)(rbase + v) * Mt + col] -= acc[tr][v];
  }
}

// ---------------------------------------------------------------------------
// 7. V = L^{-1} vec(Y^T): one workgroup, parallel dot + tree reduction.
// ---------------------------------------------------------------------------
__global__ void k_solve_v(const float* __restrict__ K, const float* __restrict__ Y,
                          float* __restrict__ V, int M, int N, int l) {
  __shared__ float v[4096];                   // M <= 4096
  __shared__ float red[256];
  int t = threadIdx.x;
  for (int idx = t; idx < M; idx += 256) {
    int task = idx / N, n = idx % N;
    v[idx] = Y[(size_t)n * l + task];         // vec of Y^T
  }
  __syncthreads();
  for (int k = 0; k < M; ++k) {
    float partial = 0.f;
    const float* row = K + (size_t)k * M;
    for (int p = t; p < k; p += 256) partial += row[p] * v[p];
    red[t] = partial;
    __syncthreads();
    for (int s = 128; s > 0; s >>= 1) {
      if (t < s) red[t] += red[t + s];
      __syncthreads();
    }
    if (t == 0) v[k] = (v[k] - red[0]) / row[k];
    __syncthreads();
  }
  for (int idx = t; idx < M; idx += 256) V[idx] = v[idx];
}

// ---------------------------------------------------------------------------
// 8. fmean = A^T V; also the (N_, l) reshape output.
// ---------------------------------------------------------------------------
__global__ void k_fmean(const float* __restrict__ A, const float* __restrict__ V,
                        float* __restrict__ fmean, float* __restrict__ out4,
                        int M, int Mt, int N_, int l) {
  __shared__ float vs[256];
  int c = blockIdx.x * 256 + threadIdx.x;
  float acc = 0.f;
  for (int kb = 0; kb < M; kb += 256) {
    __syncthreads();
    vs[threadIdx.x] = V[kb + threadIdx.x];
    __syncthreads();
    for (int kk = 0; kk < 256; ++kk)
      acc += A[(size_t)(kb + kk) * Mt + c] * vs[kk];
  }
  if (c < Mt) {
    fmean[c] = acc;
    int task = c / N_, n = c % N_;
    out4[(size_t)n * l + task] = acc;         // fmean.reshape(l, N_).T
  }
}

// ---------------------------------------------------------------------------
// Host-side orchestration (all launches on `stream`, graph-capture safe).
// ---------------------------------------------------------------------------
extern "C" void kernel_launch(void* const* d_in, const int* in_sizes, int n_in,
                              void* d_out, int out_size, void* d_ws, size_t ws_size,
                              hipStream_t stream) {
  const float* X       = (const float*)d_in[0];
  const float* test_X  = (const float*)d_in[1];
  const float* Y       = (const float*)d_in[2];
  const float* log_ns  = (const float*)d_in[3];
  const float* cf      = (const float*)d_in[4];
  const float* log_var = (const float*)d_in[5];
  const float* log_ls  = (const float*)d_in[6];

  const int l    = in_sizes[3];
  const int N    = in_sizes[2] / l;
  const int D    = in_sizes[0] / N;
  const int N_   = in_sizes[1] / D;
  const int rank = in_sizes[4] / l;
  const int M    = N * l;      // train side (4096)
  const int Mt   = N_ * l;     // test side (4096)
  const int nb   = M / BLK;

  // workspace layout (floats)
  float* ws = (float*)d_ws;
  float* wsK        = ws;                           size_t off = (size_t)M * M;
  float* wsA        = ws + off;                     off += (size_t)M * Mt;
  float* rbf_train  = ws + off;                     off += (size_t)N * N;
  float* rbf_cross  = ws + off;                     off += (size_t)N * N_;
  float* rbf_test   = ws + off;                     off += (size_t)N_ * N_;
  float* taskK      = ws + off;                     off += 64;
  float* wsV        = ws + off;

  // output layout: fmean | fvar | noise | fmean.reshape(l,N_).T
  float* out_fmean = (float*)d_out;
  float* out_fvar  = out_fmean + Mt;
  float* out_noise = out_fvar + (size_t)Mt * Mt;
  float* out_4     = out_noise + (size_t)M * M;

  // --- small precomputes -------------------------------------------------
  k_task<<<1, 64, 0, stream>>>(cf, log_var, taskK, l, rank);
  k_rbf<<<dim3((N + 255) / 256, N),   256, 0, stream>>>(X, X, rbf_train, N, N, D, log_ls);
  k_rbf<<<dim3((N_ + 255) / 256, N),  256, 0, stream>>>(X, test_X, rbf_cross, N, N_, D, log_ls);
  k_rbf<<<dim3((N_ + 255) / 256, N_), 256, 0, stream>>>(test_X, test_X, rbf_test, N_, N_, D, log_ls);

  // --- assemble K (+noise output), Kx, Kxx -------------------------------
  k_assemble_K<<<dim3((M + 255) / 256, M), 256, 0, stream>>>(taskK, rbf_train, log_ns,
                                                             wsK, out_noise, N, l);
  k_kron<<<dim3((Mt + 255) / 256, M), 256, 0, stream>>>(taskK, rbf_cross, wsA, N, N_, l);
  k_kron<<<dim3((Mt + 255) / 256, Mt), 256, 0, stream>>>(taskK, rbf_test, out_fvar, N_, N_, l);

  // --- blocked Cholesky on wsK (lower) -----------------------------------
  for (int kb = 0; kb < nb; ++kb) {
    k_potrf<<<1, BLK, 0, stream>>>(wsK, M, kb);
    int T = nb - kb - 1;
    if (T > 0) {
      k_trsm_right<<<T, BLK, 0, stream>>>(wsK, M, kb);
      int pairs = T * (T + 1) / 2;
      k_syrk_chol<<<dim3(2, pairs), 256, 0, stream>>>(wsK, M, kb, nb);
    }
  }

  // --- blocked TRSM: wsA := L^{-1} wsA -----------------------------------
  for (int ib = 0; ib < nb; ++ib) {
    if (ib > 0) {
      k_gemm_nn_sub<<<Mt / 128, 256, 0, stream>>>(
          wsA + (size_t)ib * BLK * Mt, Mt,
          wsK + (size_t)ib * BLK * M, M,
          wsA, Mt, ib * BLK);
    }
    k_trsm_left<<<Mt / 128, 128, 0, stream>>>(wsK, M, ib, wsA, Mt);
  }

  // --- fvar = Kxx - A^T A (WMMA SYRK, async double-buffered LDS) ----------
  k_syrk_at_a<<<dim3(Mt / 128, Mt / 128), 256, 0, stream>>>(out_fvar, wsA, M, Mt);

  // --- fmean ---------------------------------------------------------------
  k_solve_v<<<1, 256, 0, stream>>>(wsK, Y, wsV, M, N, l);
  k_fmean<<<(Mt + 255) / 256, 256, 0, stream>>>(wsA, wsV, out_fmean, out_4, M, Mt, N_, l);
}